// GAT_61881888801069
// MI455X (gfx1250) — compile-verified
//
#include <hip/hip_runtime.h>
#include <stdint.h>

// ---------------------------------------------------------------------------
// GATv2 x2 layers for MI455X (gfx1250, wave32).
// GEMMs use V_WMMA_F32_16X16X4_F32 (full fp32), fully unrolled over K so all
// loads use immediate offsets and the scheduler can pipeline loads vs WMMA.
// Edge phase is warp-per-edge with coalesced float2 row access and
// non-returning f32 atomics.
// ---------------------------------------------------------------------------

typedef __attribute__((ext_vector_type(2))) float v2f;
typedef __attribute__((ext_vector_type(8))) float v8f;

#define NEG_SLOPE 0.2f
// f2ord(-inf) == ~0xFF800000 == 0x007FFFFF
#define ORD_NEG_INF 0x007FFFFFu

__device__ __forceinline__ unsigned f2ord(float f) {
    unsigned u = __float_as_uint(f);
    return (u & 0x80000000u) ? ~u : (u | 0x80000000u);
}
__device__ __forceinline__ float ord2f(unsigned u) {
    unsigned b = (u & 0x80000000u) ? (u & 0x7fffffffu) : ~u;
    return __uint_as_float(b);
}

// ---------------------------------------------------------------------------
// Fused 3-matrix GEMM: Y{0,1,2}[N x 64] = X[N x K] @ W{0,1,2}[K x 64] + b{0,1,2}
// One wave (32 lanes) per 16x16 output tile, accumulating via
// v_wmma_f32_16x16x4_f32 over K in steps of 4 (fully unrolled, K is
// compile-time so all loads carry immediate offsets).
//
// OOB note: rows >= N are clamped to row N-1 and loaded unconditionally;
// garbage in A row M only affects D row M, and those rows are never stored.
//
//   grid = (ceil(N/16), 3 * (64/16)), block = 32
// ---------------------------------------------------------------------------
template <int K>
__global__ __launch_bounds__(32)
void gat_gemm3_wmma(const float* __restrict__ X,
                    const float* __restrict__ W0, const float* __restrict__ b0, float* __restrict__ Y0,
                    const float* __restrict__ W1, const float* __restrict__ b1, float* __restrict__ Y1,
                    const float* __restrict__ W2, const float* __restrict__ b2, float* __restrict__ Y2,
                    int N)
{
    constexpr int C = 64;

    const int l     = threadIdx.x;       // 0..31
    const int mrow  = l & 15;            // lane's row/col within tile
    const int khalf = l >> 4;            // 0 or 1

    const int row0 = blockIdx.x * 16;
    const int mat  = blockIdx.y >> 2;          // 0..2
    const int col0 = (blockIdx.y & 3) << 4;    // 0,16,32,48

    const float* W    = (mat == 0) ? W0 : ((mat == 1) ? W1 : W2);
    const float* bias = (mat == 0) ? b0 : ((mat == 1) ? b1 : b2);
    float*       Y    = (mat == 0) ? Y0 : ((mat == 1) ? Y1 : Y2);

    // Clamp row (unconditional loads; OOB rows never stored).
    int arow = row0 + mrow;
    if (arow > N - 1) arow = N - 1;

    // A base: this lane reads x[arow][2*khalf + kk .. +1] each step.
    const float* __restrict__ xbase = X + (size_t)arow * (size_t)K + 2 * khalf;
    // B base: this lane reads W[2*khalf + kk + j][col0 + mrow].
    const float* __restrict__ wbase = W + (size_t)(2 * khalf) * C + col0 + mrow;

    v8f acc = {};
#pragma unroll
    for (int kk = 0; kk < K; kk += 4) {
        v2f a, b;
        const float2 av = *(const float2*)(xbase + kk);   // global_load_b64, imm offset
        a.x = av.x;
        a.y = av.y;
        b.x = wbase[(size_t)(kk + 0) * C];                // imm offsets off fixed base
        b.y = wbase[(size_t)(kk + 1) * C];
        acc = __builtin_amdgcn_wmma_f32_16x16x4_f32(
            /*neg_a=*/false, a, /*neg_b=*/false, b,
            /*c_mod=*/(short)0, acc, /*reuse_a=*/false, /*reuse_b=*/false);
    }

    const float bv = bias[col0 + mrow];
    float* __restrict__ ybase = Y + (size_t)(row0 + 8 * khalf) * C + col0 + mrow;
#pragma unroll
    for (int i = 0; i < 8; ++i) {
        const int r = row0 + i + 8 * khalf;   // C/D layout: lanes16-31 hold M=i+8
        if (r < N) ybase[(size_t)i * C] = acc[i] + bv;
    }
}

// ---------------------------------------------------------------------------
// Init helpers
// ---------------------------------------------------------------------------
__global__ void fill_f32(float* __restrict__ p, float v, long long n) {
    long long i = (long long)blockIdx.x * blockDim.x + threadIdx.x;
    if (i < n) p[i] = v;
}
__global__ void fill_u32(unsigned* __restrict__ p, unsigned v, long long n) {
    long long i = (long long)blockIdx.x * blockDim.x + threadIdx.x;
    if (i < n) p[i] = v;
}

// ---------------------------------------------------------------------------
// Pass 1: per-edge attention logit + segment max (warp per edge, C==64)
//   e = att . leaky_relu(xl[src] + xr[dst]);  atomicMax(m[dst], ord(e))
// ---------------------------------------------------------------------------
__global__ void gat_edge_logits(const long long* __restrict__ ei, long long E,
                                const float* __restrict__ xl, const float* __restrict__ xr,
                                const float* __restrict__ att,
                                float* __restrict__ ebuf, unsigned* __restrict__ mbuf)
{
    const long long e = (long long)blockIdx.x * (blockDim.x >> 5) + (threadIdx.x >> 5);
    const int lane = threadIdx.x & 31;
    if (e >= E) return;
    const long long s = ei[e];
    const long long d = ei[E + e];

    const float2 a = ((const float2*)(xl + (size_t)s * 64))[lane];
    const float2 b = ((const float2*)(xr + (size_t)d * 64))[lane];
    const float2 w = ((const float2*)att)[lane];
    float z0 = a.x + b.x; z0 = (z0 > 0.0f) ? z0 : NEG_SLOPE * z0;
    float z1 = a.y + b.y; z1 = (z1 > 0.0f) ? z1 : NEG_SLOPE * z1;
    float acc = z0 * w.x + z1 * w.y;

#pragma unroll
    for (int off = 16; off > 0; off >>= 1) acc += __shfl_xor(acc, off, 32);

    if (lane == 0) {
        ebuf[e] = acc;
        atomicMax(mbuf + d, f2ord(acc));
    }
}

// ---------------------------------------------------------------------------
// Pass 2: ex = exp(e - m[dst]); segment sum (thread per edge)
// ---------------------------------------------------------------------------
__global__ void gat_edge_exp(const long long* __restrict__ ei, long long E,
                             float* __restrict__ ebuf,
                             const unsigned* __restrict__ mbuf,
                             float* __restrict__ sbuf)
{
    const long long e = (long long)blockIdx.x * blockDim.x + threadIdx.x;
    if (e >= E) return;
    const long long d = ei[E + e];
    const float m = ord2f(mbuf[d]);
    const float ex = __expf(ebuf[e] - m);
    ebuf[e] = ex;
    unsafeAtomicAdd(sbuf + d, ex);   // global_atomic_add_f32, no return
}

// ---------------------------------------------------------------------------
// Pass 3: acc[dst] += (ex/s[dst]) * xl[src]   (warp per edge, 2 ch per lane)
// ---------------------------------------------------------------------------
__global__ void gat_edge_scatter(const long long* __restrict__ ei, long long E,
                                 const float* __restrict__ ebuf,
                                 const float* __restrict__ sbuf,
                                 const float* __restrict__ xl,
                                 float* __restrict__ acc)
{
    const long long e = (long long)blockIdx.x * (blockDim.x >> 5) + (threadIdx.x >> 5);
    const int lane = threadIdx.x & 31;
    if (e >= E) return;
    const long long s = ei[e];
    const long long d = ei[E + e];
    const float alpha = ebuf[e] / sbuf[d];

    const float2 v = ((const float2*)(xl + (size_t)s * 64))[lane];
    float* pd = acc + (size_t)d * 64 + 2 * lane;
    unsafeAtomicAdd(pd + 0, alpha * v.x);
    unsafeAtomicAdd(pd + 1, alpha * v.y);
}

// ---------------------------------------------------------------------------
// Node epilogue: out = maybe_relu(acc + bc[c] + lin)   (elementwise over N*64)
// ---------------------------------------------------------------------------
__global__ void gat_epilogue(const float* __restrict__ acc,
                             const float* __restrict__ bc,
                             const float* __restrict__ lin,
                             float* __restrict__ out,
                             long long total, int do_relu)
{
    const long long i = (long long)blockIdx.x * blockDim.x + threadIdx.x;
    if (i >= total) return;
    float v = acc[i] + bc[(int)(i & 63)] + lin[i];
    out[i] = do_relu ? fmaxf(v, 0.0f) : v;
}

// ---------------------------------------------------------------------------
// Host-side orchestration (graph-capture safe: only kernel launches on stream)
// ---------------------------------------------------------------------------
extern "C" void kernel_launch(void* const* d_in, const int* in_sizes, int n_in,
                              void* d_out, int out_size, void* d_ws, size_t ws_size,
                              hipStream_t stream) {
    const float*     x    = (const float*)d_in[0];
    const long long* ei   = (const long long*)d_in[1];   // int64 edge_index [2,E]
    const float*     Wl1  = (const float*)d_in[3];
    const float*     bl1  = (const float*)d_in[4];
    const float*     Wr1  = (const float*)d_in[5];
    const float*     br1  = (const float*)d_in[6];
    const float*     att1 = (const float*)d_in[7];
    const float*     bc1  = (const float*)d_in[8];
    const float*     W1   = (const float*)d_in[9];
    const float*     b1   = (const float*)d_in[10];
    const float*     Wl2  = (const float*)d_in[11];
    const float*     bl2  = (const float*)d_in[12];
    const float*     Wr2  = (const float*)d_in[13];
    const float*     br2  = (const float*)d_in[14];
    const float*     att2 = (const float*)d_in[15];
    const float*     bc2  = (const float*)d_in[16];
    const float*     W2   = (const float*)d_in[17];
    const float*     b2   = (const float*)d_in[18];

    const int       HID  = in_sizes[4];                 // 64
    const int       D_IN = in_sizes[3] / HID;           // 128
    const int       N    = in_sizes[0] / D_IN;          // 100000
    const long long E    = (long long)in_sizes[1] / 2;  // 1600000
    const long long NC   = (long long)N * 64;

    // Workspace layout (all fp32 / u32, 4-byte aligned)
    float*    A  = (float*)d_ws;        // xl  [N*64]
    float*    B  = A + NC;              // xr  [N*64]
    float*    Cb = B + NC;              // lin1 -> h -> acc2  [N*64]
    float*    D  = Cb + NC;             // acc1 -> lin2       [N*64]
    float*    EB = D + NC;              // per-edge logits / ex [E]
    unsigned* M  = (unsigned*)(EB + E); // per-node running max (ordered) [N]
    float*    S  = (float*)(M + N);     // per-node exp-sum [N]

    const dim3 gemmGrid((unsigned)((N + 15) / 16), 12);
    const dim3 gemmBlk(32);
    const dim3 edgeWBlk(256);                          // 8 warps -> 8 edges/block
    const unsigned gEwarp = (unsigned)((E + 7) / 8);
    const unsigned gE     = (unsigned)((E + 255) / 256);
    const unsigned gNC    = (unsigned)((NC + 255) / 256);
    const unsigned gNn    = (unsigned)((N + 255) / 256);

    // ---------------- Layer 1 ----------------
    // xl1 -> A, xr1 -> B, lin1 -> C
    if (D_IN == 128) {
        gat_gemm3_wmma<128><<<gemmGrid, gemmBlk, 0, stream>>>(
            x, Wl1, bl1, A, Wr1, br1, B, W1, b1, Cb, N);
    } else {
        gat_gemm3_wmma<64><<<gemmGrid, gemmBlk, 0, stream>>>(
            x, Wl1, bl1, A, Wr1, br1, B, W1, b1, Cb, N);
    }

    fill_u32<<<gNn, 256, 0, stream>>>(M, ORD_NEG_INF, N);
    fill_f32<<<gNn, 256, 0, stream>>>(S, 0.0f, N);
    fill_f32<<<gNC, 256, 0, stream>>>(D, 0.0f, NC);

    gat_edge_logits<<<gEwarp, edgeWBlk, 0, stream>>>(ei, E, A, B, att1, EB, M);
    gat_edge_exp<<<gE, 256, 0, stream>>>(ei, E, EB, M, S);
    gat_edge_scatter<<<gEwarp, edgeWBlk, 0, stream>>>(ei, E, EB, S, A, D);

    // h = relu(acc1 + bc1 + lin1), in place into C
    gat_epilogue<<<gNC, 256, 0, stream>>>(D, bc1, Cb, Cb, NC, 1);

    // ---------------- Layer 2 ----------------
    // xl2 -> A, xr2 -> B, lin2 -> D (from h in C)
    gat_gemm3_wmma<64><<<gemmGrid, gemmBlk, 0, stream>>>(
        Cb, Wl2, bl2, A, Wr2, br2, B, W2, b2, D, N);

    fill_u32<<<gNn, 256, 0, stream>>>(M, ORD_NEG_INF, N);
    fill_f32<<<gNn, 256, 0, stream>>>(S, 0.0f, N);
    fill_f32<<<gNC, 256, 0, stream>>>(Cb, 0.0f, NC);   // acc2 (h no longer needed)

    gat_edge_logits<<<gEwarp, edgeWBlk, 0, stream>>>(ei, E, A, B, att2, EB, M);
    gat_edge_exp<<<gE, 256, 0, stream>>>(ei, E, EB, M, S);
    gat_edge_scatter<<<gEwarp, edgeWBlk, 0, stream>>>(ei, E, EB, S, A, Cb);

    // out = acc2 + bc2 + lin2 (no relu)
    gat_epilogue<<<gNC, 256, 0, stream>>>(Cb, bc2, D, (float*)d_out, NC, 0);
}